// MetaPathGATLayer_11751030522360
// MI455X (gfx1250) — compile-verified
//
#include <hip/hip_runtime.h>
#include <math.h>

#define DFEAT 64
#define HEADS 8
#define NEG_SLOPE 0.2f

typedef __attribute__((ext_vector_type(2))) float v2f;
typedef __attribute__((ext_vector_type(8))) float v8f;

__device__ __forceinline__ float leaky(float v) {
  return v >= 0.0f ? v : NEG_SLOPE * v;
}

// ---------------------------------------------------------------------------
// Kernel 1: attention scores via V_WMMA_F32_16X16X4_F32.
// One wave computes a 16x16 f32 tile: rows = 16 nodes, cols 0..7 = dst-head
// scores (h . att[:, :64]), cols 8..15 = src-head scores (h . att[:, 64:]).
// K = 64 accumulated as 16 WMMA steps of K=4.
// ---------------------------------------------------------------------------
__global__ void gat_scores_wmma(const float* __restrict__ h,
                                const float* __restrict__ att,
                                float* __restrict__ s, int N) {
  const int lane  = threadIdx.x & 31;
  const int ntile = (N + 15) >> 4;
  const int tile  = blockIdx.x * (blockDim.x >> 5) + (threadIdx.x >> 5);
  if (tile >= ntile) return;                 // wave-uniform: EXEC stays all-1s
  const int m    = lane & 15;
  const int kgrp = lane >> 4;                // 0 or 1
  int node  = tile * 16 + m;
  int nodeC = node < N ? node : N - 1;       // clamp, no divergence pre-WMMA
  const float* hrow = h + (size_t)nodeC * DFEAT;
  // column m of the packed 64x16 B matrix
  const float* attcol = (m < HEADS)
                          ? (att + (size_t)m * (2 * DFEAT))
                          : (att + (size_t)(m - HEADS) * (2 * DFEAT) + DFEAT);
  v8f c = {};
#pragma unroll
  for (int kk = 0; kk < DFEAT / 4; ++kk) {
    const int k0 = kk * 4 + kgrp * 2;
    v2f a, b;
    a.x = hrow[k0];   a.y = hrow[k0 + 1];
    b.x = attcol[k0]; b.y = attcol[k0 + 1];
    c = __builtin_amdgcn_wmma_f32_16x16x4_f32(false, a, false, b, (short)0, c,
                                              false, false);
  }
#pragma unroll
  for (int r = 0; r < 8; ++r) {
    int mrow = tile * 16 + r + 8 * kgrp;
    if (mrow < N) s[(size_t)mrow * 16 + m] = c[r];
  }
}

// ---------------------------------------------------------------------------
// Kernel 2: init out=0, e_max=-inf-ish, denom=0
// ---------------------------------------------------------------------------
__global__ void gat_init(float* __restrict__ out, float* __restrict__ emax,
                         float* __restrict__ denom, long total_out, int nh) {
  long i = (long)blockIdx.x * blockDim.x + threadIdx.x;
  if (i < total_out) out[i] = 0.0f;
  if (i < nh) { emax[i] = -3.0e38f; denom[i] = 0.0f; }
}

// edge logits for all 8 heads, vectorized b128 loads
__device__ __forceinline__ void edge_logits(const float* __restrict__ s,
                                            int src, int dst, float* ev) {
  const float4 d0 = *(const float4*)(s + (size_t)dst * 16);
  const float4 d1 = *(const float4*)(s + (size_t)dst * 16 + 4);
  const float4 s0 = *(const float4*)(s + (size_t)src * 16 + 8);
  const float4 s1 = *(const float4*)(s + (size_t)src * 16 + 12);
  ev[0] = leaky(d0.x + s0.x); ev[1] = leaky(d0.y + s0.y);
  ev[2] = leaky(d0.z + s0.z); ev[3] = leaky(d0.w + s0.w);
  ev[4] = leaky(d1.x + s1.x); ev[5] = leaky(d1.y + s1.y);
  ev[6] = leaky(d1.z + s1.z); ev[7] = leaky(d1.w + s1.w);
}

// ---------------------------------------------------------------------------
// Kernel 3: segment max over dst via native float atomic max
// ---------------------------------------------------------------------------
__global__ void gat_edge_max(const int* __restrict__ ei,
                             const float* __restrict__ s,
                             float* __restrict__ emax, int E) {
  int e = blockIdx.x * blockDim.x + threadIdx.x;
  if (e >= E) return;
  int src = ei[e], dst = ei[E + e];
  float ev[8];
  edge_logits(s, src, dst, ev);
#pragma unroll
  for (int hh = 0; hh < HEADS; ++hh)
    unsafeAtomicMax(&emax[(size_t)dst * HEADS + hh], ev[hh]);
}

// ---------------------------------------------------------------------------
// Kernel 4: segment sum of exp(e - max) over dst
// ---------------------------------------------------------------------------
__global__ void gat_edge_sum(const int* __restrict__ ei,
                             const float* __restrict__ s,
                             const float* __restrict__ emax,
                             float* __restrict__ denom, int E) {
  int e = blockIdx.x * blockDim.x + threadIdx.x;
  if (e >= E) return;
  int src = ei[e], dst = ei[E + e];
  float ev[8];
  edge_logits(s, src, dst, ev);
  const float4 m0 = *(const float4*)(emax + (size_t)dst * HEADS);
  const float4 m1 = *(const float4*)(emax + (size_t)dst * HEADS + 4);
  const float mv[8] = {m0.x, m0.y, m0.z, m0.w, m1.x, m1.y, m1.z, m1.w};
#pragma unroll
  for (int hh = 0; hh < HEADS; ++hh)
    unsafeAtomicAdd(&denom[(size_t)dst * HEADS + hh], __expf(ev[hh] - mv[hh]));
}

// ---------------------------------------------------------------------------
// Kernel 5: one wave32 per edge. Each lane computes alpha for head (lane&7)
// only (1 exp/lane instead of 8), then broadcasts via __shfl (lane permute).
// Lane l owns feature columns l and l+32; per head two coalesced
// global_atomic_add_f32 (consecutive lanes -> consecutive dwords in L2).
// ---------------------------------------------------------------------------
__global__ void gat_aggregate(const float* __restrict__ h,
                              const int* __restrict__ ei,
                              const float* __restrict__ s,
                              const float* __restrict__ emax,
                              const float* __restrict__ denom,
                              float* __restrict__ out, int E) {
  int wave = blockIdx.x * (blockDim.x >> 5) + (threadIdx.x >> 5);
  int lane = threadIdx.x & 31;
  if (wave >= E) return;                       // wave-uniform
  int src = ei[wave], dst = ei[E + wave];
  float x0 = h[(size_t)src * DFEAT + lane];
  float x1 = h[(size_t)src * DFEAT + 32 + lane];
  // each lane handles one head's softmax weight
  const int hh0 = lane & 7;
  float v = leaky(s[(size_t)dst * 16 + hh0] + s[(size_t)src * 16 + 8 + hh0]);
  float a_mine = __expf(v - emax[(size_t)dst * HEADS + hh0]) /
                 denom[(size_t)dst * HEADS + hh0];
  float* od = out + (size_t)dst * (HEADS * DFEAT);
#pragma unroll
  for (int hh = 0; hh < HEADS; ++hh) {
    float a = __shfl(a_mine, hh, 32);          // broadcast head hh's alpha
    unsafeAtomicAdd(&od[hh * DFEAT + lane],      x0 * a);
    unsafeAtomicAdd(&od[hh * DFEAT + 32 + lane], x1 * a);
  }
}

// ---------------------------------------------------------------------------
// Kernel 6: ELU, vectorized float4
// ---------------------------------------------------------------------------
__global__ void gat_elu(float4* __restrict__ out, long n4) {
  long i = (long)blockIdx.x * blockDim.x + threadIdx.x;
  if (i >= n4) return;
  float4 v = out[i];
  v.x = v.x > 0.0f ? v.x : expm1f(v.x);
  v.y = v.y > 0.0f ? v.y : expm1f(v.y);
  v.z = v.z > 0.0f ? v.z : expm1f(v.z);
  v.w = v.w > 0.0f ? v.w : expm1f(v.w);
  out[i] = v;
}

extern "C" void kernel_launch(void* const* d_in, const int* in_sizes, int n_in,
                              void* d_out, int out_size, void* d_ws, size_t ws_size,
                              hipStream_t stream) {
  const float* h   = (const float*)d_in[0];   // [N, 64]
  const int*   ei  = (const int*)d_in[1];     // [2, E] row0=src, row1=dst
  const float* att = (const float*)d_in[2];   // [8, 128]
  float* out = (float*)d_out;                 // [N, 512]

  const int N = in_sizes[0] / DFEAT;
  const int E = in_sizes[1] / 2;

  // workspace layout: s [N,16] | emax [N,8] | denom [N,8]  (~6.4 MB)
  float* s     = (float*)d_ws;
  float* emax  = s + (size_t)N * 16;
  float* denom = emax + (size_t)N * HEADS;

  // 1) attention scores (WMMA f32)
  {
    int ntile = (N + 15) / 16;
    int wavesPerBlock = 8;                    // 256 threads = 8 wave32
    int blocks = (ntile + wavesPerBlock - 1) / wavesPerBlock;
    gat_scores_wmma<<<blocks, 256, 0, stream>>>(h, att, s, N);
  }
  // 2) init accumulators + output
  {
    long total = (long)N * HEADS * DFEAT;
    int blocks = (int)((total + 255) / 256);
    gat_init<<<blocks, 256, 0, stream>>>(out, emax, denom, total, N * HEADS);
  }
  // 3) segment max
  gat_edge_max<<<(E + 255) / 256, 256, 0, stream>>>(ei, s, emax, E);
  // 4) segment sum of exp
  gat_edge_sum<<<(E + 255) / 256, 256, 0, stream>>>(ei, s, emax, denom, E);
  // 5) weighted scatter-add (one wave per edge)
  gat_aggregate<<<(E + 7) / 8, 256, 0, stream>>>(h, ei, s, emax, denom, out, E);
  // 6) ELU
  {
    long n4 = (long)N * HEADS * DFEAT / 4;
    gat_elu<<<(int)((n4 + 255) / 256), 256, 0, stream>>>((float4*)out, n4);
  }
}